// MultiTeacherDKMLayer_59519656788143
// MI455X (gfx1250) — compile-verified
//
#include <hip/hip_runtime.h>
#include <math.h>

#define NROWS 262144
#define DIM   128
#define KC    16
#define NT    4
#define TMAX  10
#define XP    132            // padded LDS row pitch (floats)

#define TAU_INV  1000.0f     // 1/1e-3
#define BETA_SEM 5.0f
#define EPSV     1e-8f

typedef __attribute__((ext_vector_type(2))) float v2f;
typedef __attribute__((ext_vector_type(8))) float v8f;

// D = A(16x4) * B(4x16) + C, full fp32 on the matrix pipe.
__device__ __forceinline__ v8f wmma4(v2f a, v2f b, v8f c) {
  return __builtin_amdgcn_wmma_f32_16x16x4_f32(false, a, false, b, (short)0, c,
                                               false, false);
}

// Cross-lane LDS visibility within one wave (stores -> reads, no block barrier).
__device__ __forceinline__ void lds_fence() {
  asm volatile("s_wait_dscnt 0" ::: "memory");
}

// Async global->LDS copy of one 16x128 f32 tile (row pitch 128 in memory,
// XP in LDS). 16 issues x (32 lanes x 16B) = 8KB, tracked by ASYNCcnt.
__device__ __forceinline__ void async_tile_load(const float* gsrc,
                                                const float* ldst, int lane) {
  unsigned long long ga =
      (unsigned long long)(uintptr_t)gsrc + (unsigned)(lane * 16);
  unsigned int la = (unsigned int)(uintptr_t)ldst + (unsigned)(lane * 16);
  #pragma unroll
  for (int r = 0; r < 16; ++r) {
    asm volatile("global_load_async_to_lds_b128 %0, %1, off"
                 :: "v"(la), "v"(ga) : "memory");
    ga += DIM * 4;
    la += XP * 4;
  }
}

__device__ __forceinline__ float hmax16(float v) {
  v = fmaxf(v, __shfl_xor(v, 1, 16));
  v = fmaxf(v, __shfl_xor(v, 2, 16));
  v = fmaxf(v, __shfl_xor(v, 4, 16));
  v = fmaxf(v, __shfl_xor(v, 8, 16));
  return v;
}
__device__ __forceinline__ float hsum16(float v) {
  v += __shfl_xor(v, 1, 16);
  v += __shfl_xor(v, 2, 16);
  v += __shfl_xor(v, 4, 16);
  v += __shfl_xor(v, 8, 16);
  return v;
}

// ---------------------------------------------------------------------------
// Zero/Init: clear accumulators, copy C_init, compute ||c||^2 and ||tc||^2.
__global__ void __launch_bounds__(256) k_init(
    const float* __restrict__ Cinit, const float* __restrict__ Tc,
    float* __restrict__ Cw, float* __restrict__ csq, float* __restrict__ meanb,
    float* __restrict__ tcsq, float* __restrict__ tpsum,
    float* __restrict__ ATX, float* __restrict__ colsum, float* __restrict__ inter)
{
  const int tid = threadIdx.x;
  for (int i = tid; i < KC * DIM; i += 256) { Cw[i] = Cinit[i]; ATX[i] = 0.f; }
  for (int i = tid; i < NT * KC * KC; i += 256) inter[i] = 0.f;
  if (tid < DIM) meanb[tid] = 0.f;
  if (tid < NT * KC) tpsum[tid] = 0.f;
  if (tid < KC) colsum[tid] = 0.f;
  __syncthreads();
  if (tid < NT * KC) {
    float s = 0.f;
    const float* tr = &Tc[tid * DIM];
    for (int d = 0; d < DIM; ++d) s += tr[d] * tr[d];
    tcsq[tid] = s;
  }
  if (tid < KC) {
    float s = 0.f;
    for (int d = 0; d < DIM; ++d) { float v = Cinit[tid * DIM + d]; s += v * v; }
    csq[tid] = s;
  }
}

// ---------------------------------------------------------------------------
// Column sums of X (divided by N later when centering).
__global__ void __launch_bounds__(128) k_mean(const float* __restrict__ X,
                                              float* __restrict__ meanb)
{
  const int d = threadIdx.x;
  float s = 0.f;
  for (int n = blockIdx.x; n < NROWS; n += gridDim.x) s += X[n * DIM + d];
  atomicAdd(&meanb[d], s);
}

// ---------------------------------------------------------------------------
// Center X -> Xc, and compute teacher soft assignments tp[t][n][k] (+ tp_sum).
// One wave per 16-row tile; fp32 WMMA for the X·Tc^T dots.
__global__ void __launch_bounds__(128) k_center(
    const float* __restrict__ X, const float* __restrict__ Tc,
    const float* __restrict__ meanb, const float* __restrict__ tcsq,
    float* __restrict__ Xc, float* __restrict__ tp, float* __restrict__ tpsum)
{
  __shared__ float x_lds[4][16 * XP];
  const int tid = threadIdx.x, wave = tid >> 5, lane = tid & 31;
  const int r16 = lane & 15, hi = lane >> 4;
  float* xw = x_lds[wave];
  const float invN = 1.0f / (float)NROWS;
  const v8f vz = {0.f, 0.f, 0.f, 0.f, 0.f, 0.f, 0.f, 0.f};
  float tps_acc[NT] = {0.f, 0.f, 0.f, 0.f};

  const int nwaves = gridDim.x * 4;
  for (int tile = blockIdx.x * 4 + wave; tile < NROWS / 16; tile += nwaves) {
    const int base = tile * 16;
    #pragma unroll
    for (int r = 0; r < 16; ++r) {
      float4 v = *(const float4*)&X[(base + r) * DIM + lane * 4];
      float4 m = *(const float4*)&meanb[lane * 4];
      v.x -= m.x * invN; v.y -= m.y * invN; v.z -= m.z * invN; v.w -= m.w * invN;
      *(float4*)&xw[r * XP + lane * 4] = v;
      *(float4*)&Xc[(base + r) * DIM + lane * 4] = v;
    }
    lds_fence();
    // row norms: lane covers half a row, fold halves with xor16
    float xs = 0.f;
    for (int c = 0; c < 64; ++c) { float v = xw[r16 * XP + hi * 64 + c]; xs += v * v; }
    xs += __shfl_xor(xs, 16, 32);   // lane j (and j+16) hold xsq[j&15]

    #pragma unroll
    for (int t = 0; t < NT; ++t) {
      v8f dot = vz;
      #pragma unroll
      for (int kk = 0; kk < 32; ++kk) {
        const int d0 = kk * 4 + 2 * hi;
        v2f a = { xw[r16 * XP + d0], xw[r16 * XP + d0 + 1] };
        v2f b = *(const v2f*)&Tc[(t * KC + r16) * DIM + d0];   // 32KB, cache-resident
        dot = wmma4(a, b, dot);
      }
      const float tcs = tcsq[t * KC + r16];
      float pv[8];
      #pragma unroll
      for (int r = 0; r < 8; ++r) {
        const float xsR = __shfl(xs, r + 8 * hi, 32);          // xsq of this row
        const float d2 = fmaxf(xsR + tcs - 2.0f * dot[r], 0.0f);
        pv[r] = -BETA_SEM * sqrtf(d2);
      }
      #pragma unroll
      for (int r = 0; r < 8; ++r) {     // stable softmax over the 16 lanes (clusters)
        float m = hmax16(pv[r]);
        float e = __expf(pv[r] - m);
        float s = hsum16(e);
        pv[r] = e / s;
      }
      #pragma unroll
      for (int r = 0; r < 8; ++r) {
        const int R = r + 8 * hi;
        tp[(t * NROWS + base + R) * KC + r16] = pv[r];
        tps_acc[t] += pv[r];
      }
    }
  }
  #pragma unroll
  for (int t = 0; t < NT; ++t) {
    float v = tps_acc[t];
    v += __shfl_xor(v, 16, 32);
    if (lane < 16) atomicAdd(&tpsum[t * KC + lane], v);
  }
}

// ---------------------------------------------------------------------------
// Main fused EM pass: dist -> softmax A -> accumulate A^T X, tp^T A, colsum(A).
// X tiles arrive via async global->LDS copies, double-buffered per wave.
__global__ void __launch_bounds__(64) k_pass(
    const float* __restrict__ Xc, const float* __restrict__ tp,
    const float* __restrict__ Cw, const float* __restrict__ csq,
    float* __restrict__ ATX, float* __restrict__ colsum, float* __restrict__ inter,
    float* __restrict__ Aout, int writeA)
{
  __shared__ float x_lds[2][2][16 * XP];   // [wave][buffer]
  __shared__ float a_lds[2][16 * 18];
  const int tid = threadIdx.x, wave = tid >> 5, lane = tid & 31;
  const int r16 = lane & 15, hi = lane >> 4;
  float* aw = a_lds[wave];
  const float csq_l = csq[r16];
  const v8f vz = {0.f, 0.f, 0.f, 0.f, 0.f, 0.f, 0.f, 0.f};

  v8f atx[8]; v8f itx[NT];
  #pragma unroll
  for (int i = 0; i < 8; ++i) atx[i] = vz;
  #pragma unroll
  for (int i = 0; i < NT; ++i) itx[i] = vz;
  float cs_acc = 0.f;

  const int nwaves = gridDim.x * 2;
  const int wid = blockIdx.x * 2 + wave;

  int buf = 0;
  if (wid < NROWS / 16)
    async_tile_load(Xc + (size_t)wid * 16 * DIM, &x_lds[wave][0][0], lane);

  for (int tile = wid; tile < NROWS / 16; tile += nwaves) {
    const int base = tile * 16;
    const int nxt = tile + nwaves;
    if (nxt < NROWS / 16) {
      async_tile_load(Xc + (size_t)nxt * 16 * DIM, &x_lds[wave][buf ^ 1][0], lane);
      asm volatile("s_wait_asynccnt 0x10" ::: "memory");  // current tile landed
    } else {
      asm volatile("s_wait_asynccnt 0x0" ::: "memory");
    }
    const float* xw = &x_lds[wave][buf][0];

    // dist dots: X(16x128) · C^T(128x16) via 32 fp32 WMMAs
    v8f dot = vz;
    #pragma unroll
    for (int kk = 0; kk < 32; ++kk) {
      const int d0 = kk * 4 + 2 * hi;
      v2f a = { xw[r16 * XP + d0], xw[r16 * XP + d0 + 1] };
      v2f b = *(const v2f*)&Cw[r16 * DIM + d0];   // 8KB, cache-resident
      dot = wmma4(a, b, dot);
    }
    // sharp softmax; row-constant xsq cancels: logits' = (2*dot - csq)/tau
    float av[8];
    #pragma unroll
    for (int r = 0; r < 8; ++r) {
      float lg = (2.0f * dot[r] - csq_l) * TAU_INV;
      float m = hmax16(lg);
      float e = __expf(lg - m);
      float s = hsum16(e);
      av[r] = e / s;
    }
    #pragma unroll
    for (int r = 0; r < 8; ++r) {
      const int R = r + 8 * hi;
      aw[R * 18 + r16] = av[r];
      cs_acc += av[r];
      if (writeA) Aout[(base + R) * KC + r16] = av[r];
    }
    lds_fence();
    // A^T·X and tp^T·A: K-dim = rows, 4 chunks of 4; same aT float2 reused
    #pragma unroll
    for (int mm = 0; mm < 4; ++mm) {
      const int m0 = mm * 4 + 2 * hi;
      v2f aT = { aw[m0 * 18 + r16], aw[(m0 + 1) * 18 + r16] };
      #pragma unroll
      for (int dt = 0; dt < 8; ++dt) {
        v2f xb = { xw[m0 * XP + dt * 16 + r16], xw[(m0 + 1) * XP + dt * 16 + r16] };
        atx[dt] = wmma4(aT, xb, atx[dt]);
      }
      #pragma unroll
      for (int t = 0; t < NT; ++t) {
        const float* tpp = &tp[(t * NROWS + base + m0) * KC + r16];
        v2f tpa = { tpp[0], tpp[KC] };
        itx[t] = wmma4(tpa, aT, itx[t]);
      }
    }
    buf ^= 1;
  }
  // one atomic flush per wave
  #pragma unroll
  for (int dt = 0; dt < 8; ++dt)
    #pragma unroll
    for (int r = 0; r < 8; ++r)
      atomicAdd(&ATX[(r + 8 * hi) * DIM + dt * 16 + r16], atx[dt][r]);
  #pragma unroll
  for (int t = 0; t < NT; ++t)
    #pragma unroll
    for (int r = 0; r < 8; ++r)
      atomicAdd(&inter[t * 256 + (r + 8 * hi) * KC + r16], itx[t][r]);
  cs_acc += __shfl_xor(cs_acc, 16, 32);
  if (lane < 16) atomicAdd(&colsum[lane], cs_acc);
}

// ---------------------------------------------------------------------------
// Tiny matching math + centroid update; also resets accumulators for next iter.
__global__ void __launch_bounds__(256) k_finalize(
    const float* __restrict__ Tc, const float* __restrict__ alphas,
    const float* __restrict__ tpsum, const float* __restrict__ tcsq,
    float* __restrict__ ATX, float* __restrict__ colsum, float* __restrict__ inter,
    float* __restrict__ Cw, float* __restrict__ csq,
    float* __restrict__ Cout, int writeC)
{
  __shared__ float cold[KC * DIM];
  __shared__ float wlds[NT * KC * KC];
  __shared__ float rs[NT * KC];
  const int tid = threadIdx.x;
  for (int i = tid; i < KC * DIM; i += 256) cold[i] = Cw[i];
  __syncthreads();
  for (int p = tid; p < NT * KC * KC; p += 256) {
    const int t = p >> 8, i = (p >> 4) & 15, k = p & 15;
    const float iv = inter[p];
    const float un = tpsum[t * KC + i] + colsum[k] - iv + EPSV;
    const float J = iv / un;
    float dotv = 0.f;
    const float* tr = &Tc[(t * KC + i) * DIM];
    for (int d = 0; d < DIM; ++d) dotv += tr[d] * cold[k * DIM + d];
    const float d2 = fmaxf(tcsq[t * KC + i] + csq[k] - 2.0f * dotv, 0.0f);
    const float S = __expf(-sqrtf(d2));
    wlds[p] = powf(J + EPSV, 0.7f) * powf(S + EPSV, 0.3f);
  }
  __syncthreads();
  if (tid < NT * KC) {
    float s = 0.f;
    for (int k = 0; k < KC; ++k) s += wlds[tid * KC + k];
    rs[tid] = s + EPSV;
  }
  __syncthreads();
  for (int p = tid; p < KC * DIM; p += 256) {
    const int k = p >> 7, d = p & 127;
    float tm = 0.f;
    for (int t = 0; t < NT; ++t) {
      const float a = alphas[t];
      for (int i = 0; i < KC; ++i)
        tm += a * (wlds[(t * KC + i) * KC + k] / rs[t * KC + i]) * Tc[(t * KC + i) * DIM + d];
    }
    const float cn = (ATX[p] + tm) / (colsum[k] + 1.0f + EPSV);
    Cw[p] = cn;
    cold[p] = cn;
    if (writeC) Cout[p] = cn;
    ATX[p] = 0.f;
  }
  __syncthreads();
  if (tid < KC) {
    float s = 0.f;
    for (int d = 0; d < DIM; ++d) { float v = cold[tid * DIM + d]; s += v * v; }
    csq[tid] = s;
    colsum[tid] = 0.f;
  }
  for (int p = tid; p < NT * KC * KC; p += 256) inter[p] = 0.f;
}

// ---------------------------------------------------------------------------
// X_rec = A @ C (A from last E-step, C after last update), fp32 WMMA.
__global__ void __launch_bounds__(128) k_xrec(
    const float* __restrict__ A, const float* __restrict__ Cw,
    float* __restrict__ Xrec)
{
  const int tid = threadIdx.x, wave = tid >> 5, lane = tid & 31;
  const int r16 = lane & 15, hi = lane >> 4;
  const v8f vz = {0.f, 0.f, 0.f, 0.f, 0.f, 0.f, 0.f, 0.f};
  const int nwaves = gridDim.x * 4;
  for (int tile = blockIdx.x * 4 + wave; tile < NROWS / 16; tile += nwaves) {
    const int base = tile * 16;
    v8f acc[8];
    #pragma unroll
    for (int i = 0; i < 8; ++i) acc[i] = vz;
    #pragma unroll
    for (int kk = 0; kk < 4; ++kk) {
      const int k0 = kk * 4 + 2 * hi;
      v2f a = *(const v2f*)&A[(base + r16) * KC + k0];
      #pragma unroll
      for (int dt = 0; dt < 8; ++dt) {
        const int col = dt * 16 + r16;
        v2f b = { Cw[k0 * DIM + col], Cw[(k0 + 1) * DIM + col] };
        acc[dt] = wmma4(a, b, acc[dt]);
      }
    }
    #pragma unroll
    for (int dt = 0; dt < 8; ++dt)
      #pragma unroll
      for (int r = 0; r < 8; ++r)
        Xrec[(base + r + 8 * hi) * DIM + dt * 16 + r16] = acc[dt][r];
  }
}

// ---------------------------------------------------------------------------
extern "C" void kernel_launch(void* const* d_in, const int* in_sizes, int n_in,
                              void* d_out, int out_size, void* d_ws, size_t ws_size,
                              hipStream_t stream)
{
  (void)in_sizes; (void)n_in; (void)out_size; (void)ws_size;
  const float* X     = (const float*)d_in[0];
  const float* Cinit = (const float*)d_in[1];
  const float* Tc    = (const float*)d_in[2];
  const float* alph  = (const float*)d_in[3];

  float* out  = (float*)d_out;
  float* Xrec = out;                                   // (N,D)
  float* Cout = out + (size_t)NROWS * DIM;             // (K,D)
  float* Aout = Cout + KC * DIM;                       // (N,K)

  float* ws     = (float*)d_ws;
  float* tp     = ws;                                  // (T,N,K)  64 MB
  float* Cw     = tp + (size_t)NT * NROWS * KC;        // (K,D)
  float* csq    = Cw + KC * DIM;                       // (K)
  float* meanb  = csq + KC;                            // (D)
  float* tcsq   = meanb + DIM;                         // (T*K)
  float* tpsum  = tcsq + NT * KC;                      // (T*K)
  float* ATX    = tpsum + NT * KC;                     // (K,D)
  float* colsum = ATX + KC * DIM;                      // (K)
  float* inter  = colsum + KC;                         // (T,K,K)

  float* Xc = Xrec;  // reuse X_rec output region as centered-X scratch

  k_init<<<1, 256, 0, stream>>>(Cinit, Tc, Cw, csq, meanb, tcsq, tpsum,
                                ATX, colsum, inter);
  k_mean<<<1024, 128, 0, stream>>>(X, meanb);
  k_center<<<512, 128, 0, stream>>>(X, Tc, meanb, tcsq, Xc, tp, tpsum);
  for (int it = 0; it < TMAX; ++it) {
    k_pass<<<1024, 64, 0, stream>>>(Xc, tp, Cw, csq, ATX, colsum, inter,
                                    Aout, it == TMAX - 1 ? 1 : 0);
    k_finalize<<<1, 256, 0, stream>>>(Tc, alph, tpsum, tcsq, ATX, colsum, inter,
                                      Cw, csq, Cout, it == TMAX - 1 ? 1 : 0);
  }
  k_xrec<<<512, 128, 0, stream>>>(Aout, Cw, Xrec);
}